// MultiBoxLoss_15719580303863
// MI455X (gfx1250) — compile-verified
//
#include <hip/hip_runtime.h>
#include <hip/hip_bf16.h>
#include <stdint.h>

#define THRESH 0.5f
#define VAR0   0.1f
#define VAR1   0.2f

// ---- CDNA5 async global->LDS copy (ASYNCcnt path, gfx1250) -----------------
__device__ __forceinline__ void async_dword_to_lds(void* lds_ptr, const void* gptr) {
  uint32_t lds_off = (uint32_t)(uintptr_t)lds_ptr;              // low 32 bits = LDS offset
  unsigned long long ga = (unsigned long long)(uintptr_t)gptr;  // 64-bit vaddr pair
  asm volatile("global_load_async_to_lds_b32 %0, %1, off"
               :: "v"(lds_off), "v"(ga) : "memory");
}
__device__ __forceinline__ void wait_async_all() {
  asm volatile("s_wait_asynccnt 0" ::: "memory");
}

// One workgroup (256 threads, 8 wave32) per batch image.
__global__ void __launch_bounds__(256)
multibox_match_kernel(const float* __restrict__ default_boxes,  // [P,4] cx,cy,w,h
                      const float* __restrict__ targets,        // [B,O,5] x0,y0,x1,y1,label
                      float* __restrict__ out_loc,              // [B,P,4]
                      float* __restrict__ out_conf,             // [B,P]
                      int P, int O)
{
  extern __shared__ unsigned char smem_raw[];
  // layout (8-byte aligned first):
  unsigned long long* bestPrior = (unsigned long long*)smem_raw;   // [O]  packed (iou|~p)
  float* pBto  = (float*)(bestPrior + O);                          // [P]  best truth overlap
  float* stage = pBto + P;                                         // [O*5] raw target rows
  float* tbox  = stage + O * 5;                                    // [O*4] truth corners
  float* tarea = tbox + O * 4;                                     // [O]
  float* tlab  = tarea + O;                                        // [O]
  int*   bpidx = (int*)(tlab + O);                                 // [O]  best prior index
  unsigned char* pBti = (unsigned char*)(bpidx + O);               // [P]  best truth index

  const int tid = threadIdx.x;
  const int b   = blockIdx.x;

  // ---- stage this batch's targets into LDS via async DMA -------------------
  const float* gsrc = targets + (size_t)b * (size_t)(O * 5);
  if (tid < O * 5) {
    async_dword_to_lds(stage + tid, gsrc + tid);
  }
  wait_async_all();
  __syncthreads();

  if (tid < O) {
    float x0 = stage[tid * 5 + 0], y0 = stage[tid * 5 + 1];
    float x1 = stage[tid * 5 + 2], y1 = stage[tid * 5 + 3];
    tbox[tid * 4 + 0] = x0;  tbox[tid * 4 + 1] = y0;
    tbox[tid * 4 + 2] = x1;  tbox[tid * 4 + 3] = y1;
    tarea[tid] = (x1 - x0) * (y1 - y0);
    tlab[tid]  = stage[tid * 5 + 4];
    bestPrior[tid] = 0ull;
  }
  __syncthreads();

  // ---- phase 1: IoU matrix, per-prior max over truths, per-truth LDS argmax
  const float4* dbv = (const float4*)default_boxes;
  for (int p = tid; p < P; p += blockDim.x) {
    if (p + (int)blockDim.x < P)
      __builtin_prefetch(&dbv[p + blockDim.x], 0, 0);   // global_prefetch_b8

    float4 db = dbv[p];                    // cx, cy, w, h
    float ax0 = db.x - db.z * 0.5f, ay0 = db.y - db.w * 0.5f;
    float ax1 = db.x + db.z * 0.5f, ay1 = db.y + db.w * 0.5f;
    float areaB = db.z * db.w;

    float bto = -1.0f;
    int   bti = 0;
    for (int j = 0; j < O; ++j) {
      float ix = fminf(ax1, tbox[4 * j + 2]) - fmaxf(ax0, tbox[4 * j + 0]);
      float iy = fminf(ay1, tbox[4 * j + 3]) - fmaxf(ay0, tbox[4 * j + 1]);
      ix = fmaxf(ix, 0.0f);
      iy = fmaxf(iy, 0.0f);
      float inter = ix * iy;
      float uni   = tarea[j] + areaB - inter;
      float iou   = inter * __builtin_amdgcn_rcpf(uni);   // v_rcp_f32
      if (iou > bto) { bto = iou; bti = j; }
      // per-truth argmax over all priors: iou in high bits, complemented prior
      // index in low bits => u64 max == max-IoU, smallest-p tie break.
      unsigned long long pk =
          ((unsigned long long)__float_as_uint(iou) << 32) |
          (unsigned)(0x7FFFFFFF - p);
      atomicMax(&bestPrior[j], pk);                       // ds_max_u64
    }
    pBto[p] = bto;
    pBti[p] = (unsigned char)bti;
  }
  __syncthreads();

  if (tid < O)
    bpidx[tid] = 0x7FFFFFFF - (int)(bestPrior[tid] & 0xFFFFFFFFull);
  __syncthreads();

  // ---- phase 2: force-match override + encode + store ----------------------
  float4* loc_out  = (float4*)out_loc + (size_t)b * (size_t)P;
  float*  conf_out = out_conf + (size_t)b * (size_t)P;
  for (int p = tid; p < P; p += blockDim.x) {
    float bto = pBto[p];
    int   bti = (int)pBti[p];
    for (int j = 0; j < O; ++j)            // scatter override: last match wins
      if (bpidx[j] == p) { bto = 2.0f; bti = j; }

    float mx0 = tbox[4 * bti + 0], my0 = tbox[4 * bti + 1];
    float mx1 = tbox[4 * bti + 2], my1 = tbox[4 * bti + 3];
    float confv = (bto < THRESH) ? 0.0f : (tlab[bti] + 1.0f);

    float4 db = dbv[p];
    float rw = __builtin_amdgcn_rcpf(db.z);
    float rh = __builtin_amdgcn_rcpf(db.w);
    float4 o;
    o.x = ((mx0 + mx1) * 0.5f - db.x) * rw * (1.0f / VAR0);
    o.y = ((my0 + my1) * 0.5f - db.y) * rh * (1.0f / VAR0);
    o.z = __logf((mx1 - mx0) * rw) * (1.0f / VAR1);       // v_log_f32
    o.w = __logf((my1 - my0) * rh) * (1.0f / VAR1);
    loc_out[p]  = o;                                       // global_store_b128
    conf_out[p] = confv;
  }
}

extern "C" void kernel_launch(void* const* d_in, const int* in_sizes, int n_in,
                              void* d_out, int out_size, void* d_ws, size_t ws_size,
                              hipStream_t stream) {
  (void)n_in; (void)out_size; (void)d_ws; (void)ws_size;
  const float* default_boxes = (const float*)d_in[2];
  const float* targets       = (const float*)d_in[3];

  const int P = in_sizes[2] / 4;                 // 8732
  const int B = in_sizes[0] / (4 * P);           // 256
  const int O = in_sizes[3] / (B * 5);           // 50

  float* out_loc  = (float*)d_out;
  float* out_conf = out_loc + (size_t)B * (size_t)P * 4;

  size_t smem = sizeof(unsigned long long) * (size_t)O   // bestPrior
              + sizeof(float) * (size_t)P                // pBto
              + sizeof(float) * (size_t)O * 5            // stage
              + sizeof(float) * (size_t)O * 4            // tbox
              + sizeof(float) * (size_t)O                // tarea
              + sizeof(float) * (size_t)O                // tlab
              + sizeof(int)   * (size_t)O                // bpidx
              + (size_t)P;                               // pBti
  smem = (smem + 15) & ~(size_t)15;                      // ~46.6 KB -> 6 WGs/WGP

  multibox_match_kernel<<<dim3(B), dim3(256), smem, stream>>>(
      default_boxes, targets, out_loc, out_conf, P, O);
}